// GCN_80977313399687
// MI455X (gfx1250) — compile-verified
//
#include <hip/hip_runtime.h>
#include <math.h>

typedef float v2f __attribute__((ext_vector_type(2)));
typedef float v8f __attribute__((ext_vector_type(8)));

#define IN_DIM 128
#define KDIM   128
#define OUTC   64
#define BN_EPS 1e-5f

// ---------------- degree / normalization ----------------
__global__ void k_init_deg(float* deg, int N) {
  int i = blockIdx.x * blockDim.x + threadIdx.x;
  if (i < N) deg[i] = 1.0f;  // self-loop contributes 1
}

__global__ void k_count_deg(const int* __restrict__ ei, float* deg, int E) {
  int e = blockIdx.x * blockDim.x + threadIdx.x;
  if (e < E) atomicAdd(&deg[ei[E + e]], 1.0f);  // dst = ei[1][e]
}

__global__ void k_deg2dis(float* dis, int N) {
  int i = blockIdx.x * blockDim.x + threadIdx.x;
  if (i < N) dis[i] = rsqrtf(dis[i]);  // deg >= 1 always (self loops)
}

// ---------------- W repack into WMMA B-fragment order ----------------
// Wpack[kk][n][half][j] = W[4*kk + 2*half + j][n]
// -> lane's B fragment for k-step kk is 2 contiguous floats; wave reads one 256B chunk.
template <int Nc>
__global__ void k_packW(const float* __restrict__ W, float* __restrict__ Wp) {
  int idx = blockIdx.x * blockDim.x + threadIdx.x;  // over 128*Nc
  if (idx >= KDIM * Nc) return;
  int j    = idx & 1;
  int half = (idx >> 1) & 1;
  int n    = (idx >> 2) % Nc;
  int kk   = (idx >> 2) / Nc;
  Wp[idx] = W[(4 * kk + 2 * half + j) * Nc + n];
}

// ---------------- fp32 WMMA GEMM: C[M x Nc] = A[M x 128] @ W[128 x Nc] ----------------
// grid.x = M/(16*MT), blockDim = (32, Nc/16). Each wave: MT 16x16 tiles sharing B frags.
template <int Nc, int MT>
__global__ void k_gemm_wmma(const float* __restrict__ A, const float* __restrict__ Wp,
                            float* __restrict__ C) {
  constexpr int ROWS  = 16 * MT;
  constexpr int PITCH = 132;          // 16B-aligned rows AND conflict-free column reads
  __shared__ float sA[ROWS * PITCH];
  const int tileM = blockIdx.x * ROWS;
  const int lane  = threadIdx.x;      // 0..31 (wave32)
  const int wave  = threadIdx.y;
  constexpr int NTHR = 32 * (Nc / 16);
  const int tid = wave * 32 + lane;

  // A tile spans full contiguous rows -> one linear float4 stream into LDS
  const float4* Ap = (const float4*)(A + (size_t)tileM * KDIM);
  for (int i4 = tid; i4 < ROWS * (KDIM / 4); i4 += NTHR) {
    int r = i4 >> 5, c4 = i4 & 31;
    *(float4*)&sA[r * PITCH + c4 * 4] = Ap[i4];
  }
  __syncthreads();

  const int tileN = wave * 16;
  const int half  = lane >> 4;        // 0: K pair {0,1}; 1: K pair {2,3}
  const int l15   = lane & 15;
  const int koff  = half * 2;
  // packed-B base: one b64 load per K-step with immediate offset kk*Nc*16 bytes
  const v2f* Wb = (const v2f*)(Wp + (size_t)(tileN + l15) * 4 + half * 2);

  v8f acc[MT];
  #pragma unroll
  for (int t = 0; t < MT; ++t) acc[t] = (v8f){};

  for (int k = 0; k < KDIM; k += 4) {
    const int kk = k >> 2;
    v2f b = Wb[kk * (Nc * 2)];        // v2f stride: Nc*4 floats / 2
    v2f a[MT];
    #pragma unroll
    for (int t = 0; t < MT; ++t) {    // stage all A frags first -> pipelined ds loads
      a[t].x = sA[(t * 16 + l15) * PITCH + k + koff];
      a[t].y = sA[(t * 16 + l15) * PITCH + k + koff + 1];
    }
    #pragma unroll
    for (int t = 0; t < MT; ++t)
      acc[t] = __builtin_amdgcn_wmma_f32_16x16x4_f32(false, a[t], false, b,
                                                     (short)0, acc[t], false, false);
  }

  float* Cp = C + (size_t)tileM * Nc + tileN + l15;  // single base; stores use immediates
  #pragma unroll
  for (int t = 0; t < MT; ++t)
    #pragma unroll
    for (int j = 0; j < 8; ++j)
      Cp[(t * 16 + j + half * 8) * Nc] = acc[t][j];
}

// scalar tail for M not multiple of 16*MT (unused for N=50000=625*80, kept for safety)
__global__ void k_gemm_tail(const float* __restrict__ A, const float* __restrict__ W,
                            float* __restrict__ C, int Nc, int M0, int M) {
  int idx = blockIdx.x * blockDim.x + threadIdx.x;
  int rows = M - M0;
  if (idx >= rows * Nc) return;
  int r = M0 + idx / Nc, c = idx % Nc;
  float s = 0.f;
  for (int k = 0; k < KDIM; ++k) s += A[(size_t)r * KDIM + k] * W[(size_t)k * Nc + c];
  C[(size_t)r * Nc + c] = s;
}

// ---------------- aggregation ----------------
// init agg with the self-loop term: agg[i] = dis[i]^2 * hlin[i]  (also clears ws poison)
template <int F>
__global__ void k_selfloop_init(const float* __restrict__ hlin, const float* __restrict__ dis,
                                float* __restrict__ agg, int N) {
  size_t idx4 = (size_t)blockIdx.x * blockDim.x + threadIdx.x;
  if (idx4 < (size_t)N * (F / 4)) {
    int i = (int)(idx4 / (F / 4));
    float d = dis[i];
    float4 v = ((const float4*)hlin)[idx4];
    float4 o = make_float4(d * d * v.x, d * d * v.y, d * d * v.z, d * d * v.w);
    ((float4*)agg)[idx4] = o;
  }
}

// one wave per edge: gather h[src] (F/32 floats per lane), scale, atomic scatter to agg[dst]
template <int F>
__global__ void k_scatter(const float* __restrict__ hlin, const int* __restrict__ ei,
                          const float* __restrict__ dis, float* __restrict__ agg, int E) {
  constexpr int VPL = F / 32;
  int gtid = blockIdx.x * blockDim.x + threadIdx.x;
  int wid  = gtid >> 5;
  int lane = threadIdx.x & 31;
  int nw   = (gridDim.x * blockDim.x) >> 5;
  for (int e = wid; e < E; e += nw) {
    int s = ei[e], d = ei[E + e];
    float nrm = dis[s] * dis[d];
    const float* hp = hlin + (size_t)s * F + lane * VPL;
    float*       ap = agg  + (size_t)d * F + lane * VPL;
    float v[VPL];
    #pragma unroll
    for (int j = 0; j < VPL; ++j) v[j] = hp[j];   // contiguous -> b128/b64 gather
    #pragma unroll
    for (int j = 0; j < VPL; ++j) atomicAdd(ap + j, nrm * v[j]);  // global_atomic_add_f32
  }
}

// ---------------- BatchNorm ----------------
__global__ void k_zero(float* p, int n) {
  int i = blockIdx.x * blockDim.x + threadIdx.x;
  if (i < n) p[i] = 0.0f;
}

// stride = gridDim*blockDim is a multiple of 128 -> each thread owns a fixed column.
__global__ void k_colstats(const float* __restrict__ h, float* __restrict__ sum,
                           float* __restrict__ sumsq, int N) {
  size_t stride = (size_t)gridDim.x * blockDim.x;
  size_t start  = (size_t)blockIdx.x * blockDim.x + threadIdx.x;
  int col = (int)(start & 127);
  float s = 0.f, q = 0.f;
  size_t total = (size_t)N * 128;
  for (size_t idx = start; idx < total; idx += stride) {
    float v = h[idx];
    s += v; q += v * v;
  }
  atomicAdd(&sum[col], s);
  atomicAdd(&sumsq[col], q);
}

__global__ void k_bn_params(const float* __restrict__ sum, const float* __restrict__ sumsq,
                            const float* __restrict__ g, const float* __restrict__ beta,
                            float* scale, float* shift, int N) {
  int c = threadIdx.x;  // blockDim = 128
  float m = sum[c] / (float)N;
  float v = sumsq[c] / (float)N - m * m;  // biased var (matches jnp.var)
  float a = g[c] * rsqrtf(v + BN_EPS);
  scale[c] = a;
  shift[c] = beta[c] - a * m;  // conv bias b cancels inside BN exactly
}

__global__ void k_bnrelu(const float* __restrict__ agg, const float* __restrict__ scale,
                         const float* __restrict__ shift, float* __restrict__ out, int N) {
  size_t idx4 = (size_t)blockIdx.x * blockDim.x + threadIdx.x;
  if (idx4 < (size_t)N * 32) {
    int c = (int)(idx4 & 31) * 4;
    float4 v = ((const float4*)agg)[idx4];
    float4 o;
    o.x = fmaxf(scale[c + 0] * v.x + shift[c + 0], 0.f);
    o.y = fmaxf(scale[c + 1] * v.y + shift[c + 1], 0.f);
    o.z = fmaxf(scale[c + 2] * v.z + shift[c + 2], 0.f);
    o.w = fmaxf(scale[c + 3] * v.w + shift[c + 3], 0.f);
    ((float4*)out)[idx4] = o;
  }
}

// ---------------- log_softmax over 64 classes, one wave per row ----------------
__global__ void k_logsoftmax(const float* __restrict__ agg, const float* __restrict__ b3,
                             float* __restrict__ out, int N) {
  int lane = threadIdx.x & 31;
  int row  = blockIdx.x * (blockDim.x >> 5) + (threadIdx.x >> 5);
  if (row >= N) return;
  float v0 = agg[(size_t)row * OUTC + lane]      + b3[lane];
  float v1 = agg[(size_t)row * OUTC + lane + 32] + b3[lane + 32];
  float m = fmaxf(v0, v1);
  #pragma unroll
  for (int off = 16; off > 0; off >>= 1) m = fmaxf(m, __shfl_xor(m, off, 32));
  float s = expf(v0 - m) + expf(v1 - m);
  #pragma unroll
  for (int off = 16; off > 0; off >>= 1) s += __shfl_xor(s, off, 32);
  float l = m + logf(s);
  out[(size_t)row * OUTC + lane]      = v0 - l;
  out[(size_t)row * OUTC + lane + 32] = v1 - l;
}

// ---------------- driver ----------------
extern "C" void kernel_launch(void* const* d_in, const int* in_sizes, int n_in,
                              void* d_out, int out_size, void* d_ws, size_t ws_size,
                              hipStream_t stream) {
  const float* x   = (const float*)d_in[0];
  const int*   ei  = (const int*)  d_in[1];
  const float* W1  = (const float*)d_in[2];
  // d_in[3] = b1 : cancels in BN
  const float* g1  = (const float*)d_in[4];
  const float* be1 = (const float*)d_in[5];
  const float* W2  = (const float*)d_in[6];
  // d_in[7] = b2 : cancels in BN
  const float* g2  = (const float*)d_in[8];
  const float* be2 = (const float*)d_in[9];
  const float* W3  = (const float*)d_in[10];
  const float* b3  = (const float*)d_in[11];
  float* out = (float*)d_out;

  const int N = in_sizes[0] / IN_DIM;
  const int E = in_sizes[1] / 2;

  float* ws  = (float*)d_ws;
  float* dis = ws;                          // N (also used as deg)
  float* B0  = dis + N;                     // N*128 : layer input h
  float* B1  = B0 + (size_t)N * 128;        // N*128 : h @ W
  float* B2  = B1 + (size_t)N * 128;        // N*128 : aggregated
  float* st  = B2 + (size_t)N * 128;        // 512: sum, sumsq, scale, shift
  float *sum = st, *sumsq = st + 128, *scale = st + 256, *shift = st + 384;
  float* Wpk = st + 512;                    // 128*128 : packed weight fragments

  const int T = 256;
  constexpr int MT = 5;                     // 80 rows per block; 50000 = 625*80
  const int mtB = N / (16 * MT);
  const int M0  = mtB * 16 * MT;
  const int rem = N - M0;

  // symmetric normalization: deg -> dis = rsqrt(deg)
  k_init_deg<<<(N + T - 1) / T, T, 0, stream>>>(dis, N);
  k_count_deg<<<(E + T - 1) / T, T, 0, stream>>>(ei, dis, E);
  k_deg2dis<<<(N + T - 1) / T, T, 0, stream>>>(dis, N);

  const int nV4_128 = (int)(((size_t)N * 32 + T - 1) / T);  // float4 elems, F=128

  // ---- layer 1 ----
  k_packW<128><<<(KDIM * 128 + T - 1) / T, T, 0, stream>>>(W1, Wpk);
  k_gemm_wmma<128, MT><<<mtB, dim3(32, 8), 0, stream>>>(x, Wpk, B1);
  if (rem) k_gemm_tail<<<(rem * 128 + T - 1) / T, T, 0, stream>>>(x, W1, B1, 128, M0, N);
  k_selfloop_init<128><<<nV4_128, T, 0, stream>>>(B1, dis, B2, N);
  k_scatter<128><<<4096, T, 0, stream>>>(B1, ei, dis, B2, E);
  k_zero<<<1, 256, 0, stream>>>(sum, 256);
  k_colstats<<<1024, T, 0, stream>>>(B2, sum, sumsq, N);
  k_bn_params<<<1, 128, 0, stream>>>(sum, sumsq, g1, be1, scale, shift, N);
  k_bnrelu<<<nV4_128, T, 0, stream>>>(B2, scale, shift, B0, N);

  // ---- layer 2 ----
  k_packW<128><<<(KDIM * 128 + T - 1) / T, T, 0, stream>>>(W2, Wpk);
  k_gemm_wmma<128, MT><<<mtB, dim3(32, 8), 0, stream>>>(B0, Wpk, B1);
  if (rem) k_gemm_tail<<<(rem * 128 + T - 1) / T, T, 0, stream>>>(B0, W2, B1, 128, M0, N);
  k_selfloop_init<128><<<nV4_128, T, 0, stream>>>(B1, dis, B2, N);
  k_scatter<128><<<4096, T, 0, stream>>>(B1, ei, dis, B2, E);
  k_zero<<<1, 256, 0, stream>>>(sum, 256);
  k_colstats<<<1024, T, 0, stream>>>(B2, sum, sumsq, N);
  k_bn_params<<<1, 128, 0, stream>>>(sum, sumsq, g2, be2, scale, shift, N);
  k_bnrelu<<<nV4_128, T, 0, stream>>>(B2, scale, shift, B0, N);

  // ---- layer 3 + log_softmax ----
  k_packW<OUTC><<<(KDIM * OUTC + T - 1) / T, T, 0, stream>>>(W3, Wpk);
  k_gemm_wmma<OUTC, MT><<<mtB, dim3(32, 4), 0, stream>>>(B0, Wpk, B1);
  if (rem) k_gemm_tail<<<(rem * OUTC + T - 1) / T, T, 0, stream>>>(B0, W3, B1, OUTC, M0, N);
  const int nV4_64 = (int)(((size_t)N * (OUTC / 4) + T - 1) / T);
  k_selfloop_init<OUTC><<<nV4_64, T, 0, stream>>>(B1, dis, B2, N);
  k_scatter<OUTC><<<4096, T, 0, stream>>>(B1, ei, dis, B2, E);
  k_logsoftmax<<<(N + 7) / 8, 256, 0, stream>>>(B2, b3, out, N);
}